// GAE_Encoder_37847251813253
// MI455X (gfx1250) — compile-verified
//
#include <hip/hip_runtime.h>

// ---------------------------------------------------------------------------
// 2-layer GCN (PyG GCNConv semantics) for MI455X / gfx1250.
//   deg includes self-loop; coef = rsqrt(deg[src]) * rsqrt(deg[dst])
//   layer: h = x@W ; out[dst] += h[src]*coef (incl. self loop) ; +bias ; relu
// GEMMs use V_WMMA_F32_16X16X4_F32 (exact f32 path, wave32, 16x16 tile/wave).
// Aggregation uses L2-resident f32 hardware atomics (h/agg fit in 192MB L2).
// ---------------------------------------------------------------------------

typedef float v2f __attribute__((ext_vector_type(2)));
typedef float v8f __attribute__((ext_vector_type(8)));

#define IN_CH  32
#define HID_CH 32
#define OUT_CH 16

__device__ __forceinline__ void atom_add_f32(float* p, float v) {
  // relaxed, device scope -> global_atomic_add_f32 (no return)
  __hip_atomic_fetch_add(p, v, __ATOMIC_RELAXED, __HIP_MEMORY_SCOPE_AGENT);
}

// ---- degree / normalization ------------------------------------------------
__global__ void k_init_deg(float* __restrict__ deg, int n) {
  int i = blockIdx.x * blockDim.x + threadIdx.x;
  if (i < n) deg[i] = 1.0f;  // self-loop
}

__global__ void k_count_deg(const int* __restrict__ dst, int E,
                            float* __restrict__ deg) {
  int e = blockIdx.x * blockDim.x + threadIdx.x;
  if (e < E) atom_add_f32(&deg[dst[e]], 1.0f);
}

__global__ void k_inv_sqrt(float* __restrict__ degw, int n) {
  int i = blockIdx.x * blockDim.x + threadIdx.x;
  if (i < n) degw[i] = rsqrtf(degw[i]);  // deg >= 1 always (self-loops)
}

// ---- GEMM 1: h1[N,32] = x[N,32] @ W1[32,32]  (WMMA f32 16x16x4) -----------
__global__ void k_gemm1_wmma(const float* __restrict__ x,
                             const float* __restrict__ W1,
                             float* __restrict__ h1, int N) {
  const int lane = threadIdx.x & 31;
  const int wave = threadIdx.x >> 5;
  const int tile = blockIdx.x * (blockDim.x >> 5) + wave;
  const int tm = tile >> 1;          // row tile (16 rows)
  const int tn = tile & 1;           // col tile (2 x 16 cols)
  if (tm * 16 >= N) return;          // wave-uniform: EXEC stays all-1s
  const int r16  = lane & 15;
  const int half = lane >> 4;        // K-pair selector for A/B fragments
  const int m = tm * 16 + r16;
  const int n = tn * 16 + r16;
  v8f acc = {0.f, 0.f, 0.f, 0.f, 0.f, 0.f, 0.f, 0.f};
#pragma unroll
  for (int kb = 0; kb < IN_CH / 4; ++kb) {
    const int k = kb * 4 + half * 2;
    v2f a, b;
    a.x = x[m * IN_CH + k];
    a.y = x[m * IN_CH + k + 1];
    b.x = W1[k * HID_CH + n];
    b.y = W1[(k + 1) * HID_CH + n];
    acc = __builtin_amdgcn_wmma_f32_16x16x4_f32(
        false, a, false, b, (short)0, acc, false, false);
  }
#pragma unroll
  for (int r = 0; r < 8; ++r) {
    const int row = tm * 16 + half * 8 + r;   // D layout: VGPR r -> M=r / r+8
    h1[row * HID_CH + n] = acc[r];
  }
}

// ---- GEMM 2: h2[N,16] = relu(agg1)[N,32] @ W2[32,16] ----------------------
__global__ void k_gemm2_wmma(const float* __restrict__ agg1,
                             const float* __restrict__ W2,
                             float* __restrict__ h2, int N) {
  const int lane = threadIdx.x & 31;
  const int wave = threadIdx.x >> 5;
  const int tm = blockIdx.x * (blockDim.x >> 5) + wave;  // OUT_CH==16: 1 col tile
  if (tm * 16 >= N) return;
  const int r16  = lane & 15;
  const int half = lane >> 4;
  const int m = tm * 16 + r16;
  const int n = r16;
  v8f acc = {0.f, 0.f, 0.f, 0.f, 0.f, 0.f, 0.f, 0.f};
#pragma unroll
  for (int kb = 0; kb < HID_CH / 4; ++kb) {
    const int k = kb * 4 + half * 2;
    v2f a, b;
    a.x = fmaxf(agg1[m * HID_CH + k], 0.0f);      // fused ReLU
    a.y = fmaxf(agg1[m * HID_CH + k + 1], 0.0f);
    b.x = W2[k * OUT_CH + n];
    b.y = W2[(k + 1) * OUT_CH + n];
    acc = __builtin_amdgcn_wmma_f32_16x16x4_f32(
        false, a, false, b, (short)0, acc, false, false);
  }
#pragma unroll
  for (int r = 0; r < 8; ++r) {
    const int row = tm * 16 + half * 8 + r;
    h2[row * OUT_CH + n] = acc[r];
  }
}

// ---- init aggregation buffers: bias + self-loop term (h[i] * w[i]^2) ------
__global__ void k_self1(const float* __restrict__ h1,
                        const float* __restrict__ w,
                        const float* __restrict__ b1,
                        float* __restrict__ agg1, int total) {
  int g = blockIdx.x * blockDim.x + threadIdx.x;
  if (g >= total) return;
  int i = g >> 5;            // /32
  int c = g & 31;
  float wi = w[i];
  agg1[g] = b1[c] + h1[g] * wi * wi;
}

__global__ void k_self2(const float* __restrict__ h2,
                        const float* __restrict__ w,
                        const float* __restrict__ b2,
                        float* __restrict__ out, int total) {
  int g = blockIdx.x * blockDim.x + threadIdx.x;
  if (g >= total) return;
  int i = g >> 4;            // /16
  int c = g & 15;
  float wi = w[i];
  out[g] = b2[c] + h2[g] * wi * wi;
}

// ---- edge scatter: agg[dst] += h[src] * w[src]*w[dst] ---------------------
// 8 threads per edge (4 ch each) -> the group's float4 loads cover the full
// 128B row of h1[src] coalesced; atomics land in L2 (12.8MB buffer resident).
__global__ void k_scatter1(const int* __restrict__ src,
                           const int* __restrict__ dst,
                           const float* __restrict__ w,
                           const float* __restrict__ h1,
                           float* __restrict__ agg1, int E) {
  int t = blockIdx.x * blockDim.x + threadIdx.x;
  if (t >= E * 8) return;
  int e  = t >> 3;
  int c4 = (t & 7) << 2;
  int s = src[e];
  int d = dst[e];
  float coef = w[s] * w[d];
  const float4 v = *reinterpret_cast<const float4*>(h1 + s * HID_CH + c4);
  float* p = agg1 + d * HID_CH + c4;
  atom_add_f32(p + 0, v.x * coef);
  atom_add_f32(p + 1, v.y * coef);
  atom_add_f32(p + 2, v.z * coef);
  atom_add_f32(p + 3, v.w * coef);
}

__global__ void k_scatter2(const int* __restrict__ src,
                           const int* __restrict__ dst,
                           const float* __restrict__ w,
                           const float* __restrict__ h2,
                           float* __restrict__ out, int E) {
  int t = blockIdx.x * blockDim.x + threadIdx.x;
  if (t >= E * 4) return;
  int e  = t >> 2;
  int c4 = (t & 3) << 2;
  int s = src[e];
  int d = dst[e];
  float coef = w[s] * w[d];
  const float4 v = *reinterpret_cast<const float4*>(h2 + s * OUT_CH + c4);
  float* p = out + d * OUT_CH + c4;
  atom_add_f32(p + 0, v.x * coef);
  atom_add_f32(p + 1, v.y * coef);
  atom_add_f32(p + 2, v.z * coef);
  atom_add_f32(p + 3, v.w * coef);
}

// ---------------------------------------------------------------------------
extern "C" void kernel_launch(void* const* d_in, const int* in_sizes, int n_in,
                              void* d_out, int out_size, void* d_ws, size_t ws_size,
                              hipStream_t stream) {
  const float* x    = (const float*)d_in[0];
  const int*   edge = (const int*)d_in[1];   // [2, E] flat, row-major
  const float* W1   = (const float*)d_in[2];
  const float* b1   = (const float*)d_in[3];
  const float* W2   = (const float*)d_in[4];
  const float* b2   = (const float*)d_in[5];
  float* out = (float*)d_out;

  const int N = in_sizes[0] / IN_CH;   // 100000 (multiple of 16)
  const int E = in_sizes[1] / 2;       // 1600000
  const int* src = edge;
  const int* dst = edge + E;

  // workspace layout (floats): [deg/w: N][h1: N*32][agg1: N*32][h2: N*16]
  float* ws   = (float*)d_ws;
  float* degw = ws;
  float* h1   = degw + (size_t)N;
  float* agg1 = h1 + (size_t)N * HID_CH;
  float* h2   = agg1 + (size_t)N * HID_CH;

  const int B = 256;

  // 1) degree (with self-loop) -> inv sqrt
  k_init_deg <<<(N + B - 1) / B, B, 0, stream>>>(degw, N);
  k_count_deg<<<(E + B - 1) / B, B, 0, stream>>>(dst, E, degw);
  k_inv_sqrt <<<(N + B - 1) / B, B, 0, stream>>>(degw, N);

  // 2) layer 1: GEMM (WMMA) -> self-loop+bias init -> edge scatter
  {
    const int tiles = (N / 16) * (HID_CH / 16);      // 12500
    const int wpb = 128 / 32;                        // 4 waves/block
    k_gemm1_wmma<<<(tiles + wpb - 1) / wpb, 128, 0, stream>>>(x, W1, h1, N);
  }
  k_self1<<<((N * HID_CH) + B - 1) / B, B, 0, stream>>>(h1, degw, b1, agg1,
                                                        N * HID_CH);
  k_scatter1<<<(E * 8 + B - 1) / B, B, 0, stream>>>(src, dst, degw, h1, agg1, E);

  // 3) layer 2: ReLU fused into GEMM A-loads -> init -> scatter into d_out
  {
    const int tiles = (N / 16);                      // 6250
    const int wpb = 128 / 32;
    k_gemm2_wmma<<<(tiles + wpb - 1) / wpb, 128, 0, stream>>>(agg1, W2, h2, N);
  }
  k_self2<<<((N * OUT_CH) + B - 1) / B, B, 0, stream>>>(h2, degw, b2, out,
                                                        N * OUT_CH);
  k_scatter2<<<(E * 4 + B - 1) / B, B, 0, stream>>>(src, dst, degw, h2, out, E);
}